// CGCoupler_57724360458661
// MI455X (gfx1250) — compile-verified
//
#include <hip/hip_runtime.h>

typedef float v2f __attribute__((ext_vector_type(2)));
typedef float v8f __attribute__((ext_vector_type(8)));

#define IN_DIM 2048   // 128 channels * 16 (l,m) components, l = 0..3
#define NCOMP  16     // total (l,m) components
#define NCH    128    // channels per component
#define NKP    72     // packed K (70 valid (a1,a2) columns with l1+l2<=3, padded to mult of 4)
#define NCHUNK (NKP/4)

// l of component index a: a=0 -> l0, a=1..3 -> l1, a=4..8 -> l2, a=9..15 -> l3
__host__ __device__ constexpr int l_of(int a) { return a == 0 ? 0 : (a < 4 ? 1 : (a < 9 ? 2 : 3)); }
// number of components with l <= L (components are l-ordered prefixes)
__host__ __device__ constexpr int thresh(int L) { return L == 0 ? 1 : (L == 1 ? 4 : (L == 2 ? 9 : 16)); }

struct PackTab {
    short a1[NKP];
    short a2[NKP];
};
constexpr PackTab make_tab() {
    PackTab t{};
    int n = 0;
    for (int A1 = 0; A1 < 16; ++A1) {
        int lim = thresh(3 - l_of(A1));          // valid a2 form a prefix
        for (int A2 = 0; A2 < lim; ++A2) { t.a1[n] = (short)A1; t.a2[n] = (short)A2; ++n; }
    }
    for (int i = n; i < NKP; ++i) { t.a1[i] = 0; t.a2[i] = 0; }  // padding (W stays 0 there)
    return t;
}
constexpr PackTab TAB = make_tab();

// prefix(a1) = sum_{a<a1} thresh(3 - l_of(a)); closed form so no device table needed
__device__ __forceinline__ int pack_prefix(int a1) {
    if (a1 == 0) return 0;
    if (a1 <= 3) return 16 + 9 * (a1 - 1);
    if (a1 <= 8) return 43 + 4 * (a1 - 4);
    return 63 + (a1 - 9);
}

__global__ void zero_w_kernel(float* __restrict__ Wg) {
    int i = blockIdx.x * blockDim.x + threadIdx.x;
    if (i < NCOMP * NKP) Wg[i] = 0.0f;
}

// One CG coefficient per 128-channel group: t = k*128 has channel ns=0, so
// repid = 128*a + 0  ->  a = repid >> 7. Scatter into packed-K dense W.
__global__ void scatter_w_kernel(const float* __restrict__ cg,
                                 const int* __restrict__ r1,
                                 const int* __restrict__ r2,
                                 const int* __restrict__ ro,
                                 int nnz, float* __restrict__ Wg) {
    int k = blockIdx.x * blockDim.x + threadIdx.x;
    if (k >= nnz) return;
    int t  = k * NCH;
    int a1 = r1[t] >> 7;
    int a2 = r2[t] >> 7;
    int ao = ro[t] >> 7;
    int kp = pack_prefix(a1) + a2;               // packed column
    atomicAdd(&Wg[ao * NKP + kp], cg[t]);        // (a1,a2,ao) unique; atomic for safety
}

// Recursive template chunk loop: KC is a template parameter, so the pack-table
// lookups are frontend-evaluated constant expressions (integer literals in IR).
// No device-side TAB object, no dynamic register indexing, no cndmask ladders.
template <int KC>
__device__ __forceinline__ void chunk_rec(const float (&u)[16], const float (&v)[16],
                                          bool hi, const v2f (&afrag)[NCHUNK], v8f& acc) {
    if constexpr (KC < NCHUNK) {
        constexpr int A10 = TAB.a1[4 * KC + 0], A20 = TAB.a2[4 * KC + 0];
        constexpr int A11 = TAB.a1[4 * KC + 1], A21 = TAB.a2[4 * KC + 1];
        constexpr int A12 = TAB.a1[4 * KC + 2], A22 = TAB.a2[4 * KC + 2];
        constexpr int A13 = TAB.a1[4 * KC + 3], A23 = TAB.a2[4 * KC + 3];
        // B fragment: Z[K][n]; lanes 0-15 carry K=4KC,4KC+1; lanes 16-31 carry 4KC+2,4KC+3.
        // Select operands (not products); selects between identical registers fold away.
        float ux = hi ? u[A12] : u[A10];
        float vx = hi ? v[A22] : v[A20];
        float uy = hi ? u[A13] : u[A11];
        float vy = hi ? v[A23] : v[A21];
        v2f bfrag;
        bfrag.x = ux * vx;
        bfrag.y = uy * vy;
        acc = __builtin_amdgcn_wmma_f32_16x16x4_f32(false, afrag[KC], false, bfrag,
                                                    (short)0, acc, false, false);
        chunk_rec<KC + 1>(u, v, hi, afrag, acc);
    }
}

// out[b, ao*128+n] = sum_k W[ao][k] * x1[b,a1(k)*128+n] * x2[b,a2(k)*128+n]
// One block per batch row b; wave w handles channels n in [16w, 16w+16).
// WMMA tile: M=16 output comps (ao), N=16 channels, K accumulated 4 at a time.
__global__ __launch_bounds__(256) void cg_coupler_wmma_kernel(
        const float* __restrict__ x1, const float* __restrict__ x2,
        const float* __restrict__ Wg, float* __restrict__ out) {
    __shared__ float Wlds[NCOMP * NKP];
    const int tid = threadIdx.x;
    for (int i = tid; i < NCOMP * NKP; i += 256) Wlds[i] = Wg[i];
    __syncthreads();

    const int b    = blockIdx.x;
    const int wave = tid >> 5;
    const int lane = tid & 31;
    const bool hi  = (lane & 16) != 0;   // WMMA lane half (K/M split)
    const int n    = wave * 16 + (lane & 15);

    const float* X1 = x1 + (size_t)b * IN_DIM + n;
    const float* X2 = x2 + (size_t)b * IN_DIM + n;

    // Preload ALL A-fragments (W) from LDS into registers: 18 x v2f = 36 VGPRs.
    // A-layout for f32 16x4: VGPR0 = W[row][4kc + 2*half], VGPR1 = +1.
    const int row = lane & 15;
    const float* wb = &Wlds[row * NKP + (hi ? 2 : 0)];  // 8B aligned (NKP even)
    v2f afrag[NCHUNK];
#pragma unroll
    for (int kc = 0; kc < NCHUNK; ++kc) {
        afrag[kc].x = wb[4 * kc + 0];
        afrag[kc].y = wb[4 * kc + 1];
    }

    // per-lane: all 16 components of this channel for both inputs
    float u[16], v[16];
#pragma unroll
    for (int a = 0; a < 16; ++a) { u[a] = X1[a * NCH]; v[a] = X2[a * NCH]; }

    v8f acc = {0.f, 0.f, 0.f, 0.f, 0.f, 0.f, 0.f, 0.f};
    chunk_rec<0>(u, v, hi, afrag, acc);

    // D layout: acc[g] holds D[M = g + 8*half][N = lane&15]
    float* O = out + (size_t)b * IN_DIM + n;   // out_dim == 2048 for this metadata
#pragma unroll
    for (int g = 0; g < 8; ++g) O[(g + (hi ? 8 : 0)) * NCH] = acc[g];
}

extern "C" void kernel_launch(void* const* d_in, const int* in_sizes, int n_in,
                              void* d_out, int out_size, void* d_ws, size_t ws_size,
                              hipStream_t stream) {
    const float* x1 = (const float*)d_in[0];
    const float* x2 = (const float*)d_in[1];
    const float* cg = (const float*)d_in[2];
    const int*   r1 = (const int*)d_in[3];
    const int*   r2 = (const int*)d_in[4];
    const int*   ro = (const int*)d_in[5];
    float* Wg = (float*)d_ws;                  // 16*72 floats of scratch

    const int T   = in_sizes[2];
    const int nnz = T / NCH;
    const int B   = in_sizes[0] / IN_DIM;

    zero_w_kernel<<<(NCOMP * NKP + 255) / 256, 256, 0, stream>>>(Wg);
    scatter_w_kernel<<<(nnz + 255) / 256, 256, 0, stream>>>(cg, r1, r2, ro, nnz, Wg);
    cg_coupler_wmma_kernel<<<B, 256, 0, stream>>>(x1, x2, Wg, (float*)d_out);
}